// GumbelSampler_2482491097709
// MI455X (gfx1250) — compile-verified
//
#include <hip/hip_runtime.h>
#include <math.h>

#define TPB     512            // 16 wave32s
#define NWAVES  (TPB / 32)
#define NN      65536          // N*N
#define EPT     (NN / TPB)     // 128 elements per thread (held in VGPRs)
#define KSEL    64             // top-k / iteration count
#define ROWS    256            // B*E

typedef __attribute__((ext_vector_type(2))) float v2f;
typedef __attribute__((ext_vector_type(8))) float v8f;

// Wave32 sum reduction assisted by V_WMMA_F32_16X16X4_F32.
// A = ones(16x4) -> every row m of C holds the column sums of B.
// Summing c[0] over all 32 lanes counts every B element exactly twice
// (rows m=0 and m=8), regardless of the exact B lane->slot mapping,
// because summation is permutation invariant. So wave_sum = 0.5 * shfl-sum.
__device__ __forceinline__ float wave_sum_wmma(float p) {
    v2f a  = {1.0f, 1.0f};          // A matrix: all ones
    v2f bm = {p, 0.0f};             // B matrix: one partial per lane
    v8f c  = {};
    c = __builtin_amdgcn_wmma_f32_16x16x4_f32(
            /*neg_a=*/false, a, /*neg_b=*/false, bm,
            /*c_mod=*/(short)0, c, /*reuse_a=*/false, /*reuse_b=*/false);
    float s = c[0];
#pragma unroll
    for (int m = 16; m; m >>= 1) s += __shfl_xor(s, m, 32);
    return 0.5f * s;
}

__device__ __forceinline__ float wave_max32(float v) {
#pragma unroll
    for (int m = 16; m; m >>= 1) v = fmaxf(v, __shfl_xor(v, m, 32));
    return v;
}

extern "C" __global__ void __launch_bounds__(TPB, 1)
gumbel_topk_kernel(const float* __restrict__ scores,
                   const float* __restrict__ gumbel,
                   float* __restrict__ out) {
    extern __shared__ float lds_khot[];      // NN floats = 256 KB (CDNA5 WGP LDS)
    __shared__ float s_red[NWAVES];
    __shared__ float s_val[NWAVES];
    __shared__ int   s_idx[NWAVES];

    const int r    = blockIdx.x;             // row = b*4 + e
    const int t    = threadIdx.x;
    const int wave = t >> 5;
    const int lane = t & 31;
    const int b    = r >> 2;
    const int e    = r & 3;

    const float* __restrict__ gum = gumbel + (size_t)r * NN;            // contiguous
    const float* __restrict__ sco = scores + (size_t)b * NN * 4 + e;    // stride 4 floats
    float* __restrict__       orow = out   + (size_t)b * NN * 4 + e;

    // ---- init: flat = scores(permuted) + gumbel ; khot = 0 ----
    float flat[EPT];
#pragma unroll
    for (int i = 0; i < EPT; ++i) {
        const int j = i * TPB + t;           // coalesced within a wave
        const float sc = __builtin_nontemporal_load(&sco[(size_t)j * 4]);
        const float gu = __builtin_nontemporal_load(&gum[j]);
        flat[i] = sc + gu;
        lds_khot[j] = 0.0f;
    }
    __syncthreads();

    const float A    = 10.0f * 1.4426950408889634f;  // (1/tau) * log2(e)
    const float LN2  = 0.6931471805599453f;
    const float EPS  = 1.17549435e-38f;              // np.finfo(f32).tiny

    // initial row max
    float lmax = -INFINITY;
#pragma unroll
    for (int i = 0; i < EPT; ++i) lmax = fmaxf(lmax, flat[i]);
    lmax = wave_max32(lmax);
    if (lane == 0) s_red[wave] = lmax;
    __syncthreads();
    float m = s_red[0];
#pragma unroll
    for (int w = 1; w < NWAVES; ++w) m = fmaxf(m, s_red[w]);
    __syncthreads();

    // ---- 64 gumbel-softmax iterations ----
#pragma unroll 1
    for (int it = 0; it < KSEL; ++it) {
        // Z = sum exp((flat - m)/tau)  via exp2(flat*A - m*A)
        const float bias = m * A;
        float lsum = 0.0f;
#pragma unroll
        for (int i = 0; i < EPT; ++i)
            lsum += __builtin_amdgcn_exp2f(__builtin_fmaf(flat[i], A, -bias));
        lsum = wave_sum_wmma(lsum);                  // v_wmma-assisted wave reduce
        if (lane == 0) s_red[wave] = lsum;
        __syncthreads();
        float Z = 0.0f;
#pragma unroll
        for (int w = 0; w < NWAVES; ++w) Z += s_red[w];
        __syncthreads();
        const float lbias = bias + __builtin_amdgcn_logf(Z);  // log2(Z); onehot = exp2(flat*A - lbias)

        // khot += onehot ; flat += log(max(1 - onehot, eps)) ; track next max
        float nmax = -INFINITY;
#pragma unroll
        for (int i = 0; i < EPT; ++i) {
            const int j = i * TPB + t;
            const float onehot = __builtin_amdgcn_exp2f(__builtin_fmaf(flat[i], A, -lbias));
            lds_khot[j] += onehot;                               // private slot: plain ds RMW
            const float lg2 = __builtin_amdgcn_logf(fmaxf(1.0f - onehot, EPS));
            flat[i] = __builtin_fmaf(lg2, LN2, flat[i]);          // += ln(.)
            nmax = fmaxf(nmax, flat[i]);
        }
        nmax = wave_max32(nmax);
        if (lane == 0) s_red[wave] = nmax;
        __syncthreads();
        m = s_red[0];
#pragma unroll
        for (int w = 1; w < NWAVES; ++w) m = fmaxf(m, s_red[w]);
        __syncthreads();
    }

    // ---- top-K: copy khot into (now-dead) flat registers; LDS copy stays pristine ----
#pragma unroll
    for (int i = 0; i < EPT; ++i) flat[i] = lds_khot[i * TPB + t];

    unsigned selmask[EPT / 32];
#pragma unroll
    for (int w = 0; w < EPT / 32; ++w) selmask[w] = 0u;

#pragma unroll 1
    for (int s = 0; s < KSEL; ++s) {
        // local argmax (largest value, smallest index on ties — matches lax.top_k)
        float bv = -1.0f;                     // khot >= 0, so -1 is a safe sentinel
        int   bi = 0x7FFFFFFF;
#pragma unroll
        for (int i = 0; i < EPT; ++i) {
            const int j = i * TPB + t;
            if (flat[i] > bv || (flat[i] == bv && j < bi)) { bv = flat[i]; bi = j; }
        }
        // wave32 argmax
#pragma unroll
        for (int mm = 16; mm; mm >>= 1) {
            const float ov = __shfl_xor(bv, mm, 32);
            const int   oi = __shfl_xor(bi, mm, 32);
            if (ov > bv || (ov == bv && oi < bi)) { bv = ov; bi = oi; }
        }
        if (lane == 0) { s_val[wave] = bv; s_idx[wave] = bi; }
        __syncthreads();
        float gv = s_val[0]; int gi = s_idx[0];
#pragma unroll
        for (int w = 1; w < NWAVES; ++w) {
            const float wv = s_val[w]; const int wi = s_idx[w];
            if (wv > gv || (wv == gv && wi < gi)) { gv = wv; gi = wi; }
        }
        __syncthreads();
        // owner disables winner and records the selection bit
        if ((gi & (TPB - 1)) == t) {
            const int ii = gi / TPB;
#pragma unroll
            for (int i = 0; i < EPT; ++i) {
                if (i == ii) { flat[i] = -1.0f; selmask[i >> 5] |= (1u << (i & 31)); }
            }
        }
    }

    // ---- output: (hard - khot) + khot  ->  0.0 or (1-khot)+khot ; NT streaming stores ----
#pragma unroll
    for (int i = 0; i < EPT; ++i) {
        const int j = i * TPB + t;
        const float kh  = lds_khot[j];
        const bool  sel = (selmask[i >> 5] >> (i & 31)) & 1u;
        const float res = sel ? ((1.0f - kh) + kh) : 0.0f;
        __builtin_nontemporal_store(res, &orow[(size_t)j * 4]);
    }
}

extern "C" void kernel_launch(void* const* d_in, const int* in_sizes, int n_in,
                              void* d_out, int out_size, void* d_ws, size_t ws_size,
                              hipStream_t stream) {
    (void)in_sizes; (void)n_in; (void)d_ws; (void)ws_size; (void)out_size;
    const float* scores = (const float*)d_in[0];
    const float* gumbel = (const float*)d_in[1];
    float* out = (float*)d_out;

    const size_t shmem = (size_t)NN * sizeof(float);   // 256 KB dynamic LDS (<= 320 KB/WGP)
    static_assert(TPB * EPT == NN, "tiling");
    hipFuncSetAttribute((const void*)gumbel_topk_kernel,
                        hipFuncAttributeMaxDynamicSharedMemorySize, (int)shmem);
    hipLaunchKernelGGL(gumbel_topk_kernel, dim3(ROWS), dim3(TPB), shmem, stream,
                       scores, gumbel, out);
}